// HybridQLSTM_65481071396942
// MI455X (gfx1250) — compile-verified
//
#include <hip/hip_runtime.h>
#include <cstddef>
#include <cstdint>

typedef __attribute__((ext_vector_type(2))) float v2f;
typedef __attribute__((ext_vector_type(4))) float v4f;
typedef __attribute__((ext_vector_type(8))) float v8f;
typedef __attribute__((ext_vector_type(4))) unsigned int u32x4;
typedef __attribute__((ext_vector_type(8))) int i32x8;
typedef __attribute__((ext_vector_type(4))) int i32x4;

#define SEQ     128
#define BATCH   512
#define INDIM   512
#define NQ      8
#define NCOL    64          // 4 gates * 8 qubits * 2 used angles
#define MTOT    (SEQ*BATCH) // 65536 GEMM rows
#define BPANEL_ELEMS  (256*NCOL*2)         // 32768 floats = 128 KB

// ---------------------------------------------------------------------------
// Kernel A: pack the used weight rows.
//   col n = gate*16 + q*2 + axis  ->  source row q*3+axis of W_gate (24x520)
//   Wpack[(k/2)*64 + n] = float2{ W[row][k], W[row][k+1] }   (K-pair major, WMMA B)
//   Whpack[n*8 + j]     = W[row][512+j],   bpack[n] = b_gate[row]
// ---------------------------------------------------------------------------
__global__ __launch_bounds__(256) void pack_weights(
    const float* __restrict__ Wf, const float* __restrict__ bf,
    const float* __restrict__ Wi, const float* __restrict__ bi,
    const float* __restrict__ Wg, const float* __restrict__ bg,
    const float* __restrict__ Wo, const float* __restrict__ bo,
    float* __restrict__ Wpack, float* __restrict__ Whpack, float* __restrict__ bpack)
{
    const float* Ws[4] = {Wf, Wi, Wg, Wo};
    const float* bs[4] = {bf, bi, bg, bo};
    int tid = blockIdx.x * blockDim.x + threadIdx.x;

    if (tid < NCOL * 256) {
        int n = tid & 63, kk = tid >> 6;
        int gate = n >> 4, q = (n >> 1) & 7, axis = n & 1;
        int row = q * 3 + axis;
        const float* W = Ws[gate];
        Wpack[(kk * 64 + n) * 2 + 0] = W[row * 520 + 2 * kk + 0];
        Wpack[(kk * 64 + n) * 2 + 1] = W[row * 520 + 2 * kk + 1];
    } else if (tid < NCOL * 256 + NCOL * NQ) {
        int r = tid - NCOL * 256;
        int n = r >> 3, j = r & 7;
        int gate = n >> 4, q = (n >> 1) & 7, axis = n & 1;
        Whpack[n * 8 + j] = Ws[gate][(q * 3 + axis) * 520 + 512 + j];
    } else if (tid < NCOL * 256 + NCOL * NQ + NCOL) {
        int n = tid - (NCOL * 256 + NCOL * NQ);
        int gate = n >> 4, q = (n >> 1) & 7, axis = n & 1;
        bpack[n] = bs[gate][q * 3 + axis];
    }
}

// ---------------------------------------------------------------------------
// Kernel B: pre[t][n][b] = bias[n] + sum_k X[t*512+b][k] * Wrow_n[k]
// fp32 WMMA 16x16x4; each wave owns TWO 16-row M tiles sharing one B fetch.
// Whole B panel (128 KB) DMA'd to LDS once via TDM (tensor_load_to_lds),
// synchronized with s_wait_tensorcnt. K loop software-pipelined by 1.
// ---------------------------------------------------------------------------
__device__ __forceinline__ void tdm_load_panel(const float* gsrc, void* lds_base) {
    uint64_t ga = (uint64_t)(uintptr_t)gsrc;
    uint32_t la = (uint32_t)(uintptr_t)lds_base;      // LDS aperture: offset in low 32b
    // D# group 0: count=1 | lds_addr | global_addr[56:0] | type=2
    u32x4 g0 = { 1u, la,
                 (uint32_t)(ga & 0xFFFFFFFFu),
                 (uint32_t)((ga >> 32) & 0x01FFFFFFu) | (2u << 30) };
    // D# group 1: data_size=4B; tensor_dim0 = tile_dim0 = 32768; dim1 = 1;
    //             tensor_dim0_stride = 32768
    i32x8 g1 = { (int)0x00020000,                       // wgmask=0, data_size=2 (4B)
                 (int)((BPANEL_ELEMS & 0xFFFF) << 16),  // tensor_dim0[15:0] << 16
                 (int)(1 << 16) | (int)(BPANEL_ELEMS >> 16), // dim0[31:16] | dim1[15:0]<<16
                 (int)((BPANEL_ELEMS & 0xFFFF) << 16),  // dim1[31:16]=0 | tile_dim0<<16
                 0,                                     // tile_dim1=0, tile_dim2=0
                 (int)BPANEL_ELEMS,                     // tensor_dim0_stride[31:0]
                 0, 0 };
    i32x4 gz = { 0, 0, 0, 0 };
#if defined(__clang_major__) && (__clang_major__ >= 23)
    i32x8 gz8 = { 0, 0, 0, 0, 0, 0, 0, 0 };
    __builtin_amdgcn_tensor_load_to_lds(g0, g1, gz, gz, gz8, 0);
#else
    __builtin_amdgcn_tensor_load_to_lds(g0, g1, gz, gz, 0);
#endif
    __builtin_amdgcn_s_wait_tensorcnt(0);
}

#define WMMA_F32(a, b, c) \
    __builtin_amdgcn_wmma_f32_16x16x4_f32(false, (a), false, (b), (short)0, (c), false, false)

__global__ __launch_bounds__(256) void qgemm(
    const float* __restrict__ X,      // [65536][512]
    const float* __restrict__ Wpack,  // [256][64] float2 (K-pair major)
    const float* __restrict__ bpack,  // [64]
    float* __restrict__ preT)         // [128][64][512]
{
    extern __shared__ char smem[];                 // 128 KB dynamic LDS
    const v2f* Bs = (const v2f*)smem;

    int tid = threadIdx.x;
    int wave = tid >> 5;
    int lane = tid & 31;
    int l16 = lane & 15, lh = lane >> 4;

    if (wave == 0) tdm_load_panel(Wpack, smem);    // one DMA per workgroup
    __syncthreads();

    int pairbase = (blockIdx.x * 8 + wave) * 2;    // 2 consecutive M tiles
    int mbase = pairbase * 16;                     // multiple of 32
    int t     = mbase >> 9;
    int bbase = mbase & 511;

    v8f acc[2][4];
#pragma unroll
    for (int nt = 0; nt < 4; ++nt) {
        float bv = bpack[nt * 16 + l16];
#pragma unroll
        for (int r = 0; r < 8; ++r) { acc[0][nt][r] = bv; acc[1][nt][r] = bv; }
    }

    const float* xp0 = X + (size_t)(mbase + l16) * INDIM + 2 * lh;
    const float* xp1 = xp0 + (size_t)16 * INDIM;

    // software pipeline: regs for K-step ks while loading ks+1
    v2f a0 = *(const v2f*)(xp0);
    v2f a1 = *(const v2f*)(xp1);
    int kk0 = lh;
    v2f b0 = Bs[kk0 * 64 +  0 + l16];
    v2f b1 = Bs[kk0 * 64 + 16 + l16];
    v2f b2 = Bs[kk0 * 64 + 32 + l16];
    v2f b3 = Bs[kk0 * 64 + 48 + l16];

#pragma unroll 4
    for (int ks = 0; ks < 127; ++ks) {
        v2f a0n = *(const v2f*)(xp0 + (ks + 1) * 4);
        v2f a1n = *(const v2f*)(xp1 + (ks + 1) * 4);
        int kkn = (ks + 1) * 2 + lh;
        v2f b0n = Bs[kkn * 64 +  0 + l16];
        v2f b1n = Bs[kkn * 64 + 16 + l16];
        v2f b2n = Bs[kkn * 64 + 32 + l16];
        v2f b3n = Bs[kkn * 64 + 48 + l16];

        acc[0][0] = WMMA_F32(a0, b0, acc[0][0]);
        acc[0][1] = WMMA_F32(a0, b1, acc[0][1]);
        acc[0][2] = WMMA_F32(a0, b2, acc[0][2]);
        acc[0][3] = WMMA_F32(a0, b3, acc[0][3]);
        acc[1][0] = WMMA_F32(a1, b0, acc[1][0]);
        acc[1][1] = WMMA_F32(a1, b1, acc[1][1]);
        acc[1][2] = WMMA_F32(a1, b2, acc[1][2]);
        acc[1][3] = WMMA_F32(a1, b3, acc[1][3]);

        a0 = a0n; a1 = a1n; b0 = b0n; b1 = b1n; b2 = b2n; b3 = b3n;
    }
    acc[0][0] = WMMA_F32(a0, b0, acc[0][0]);
    acc[0][1] = WMMA_F32(a0, b1, acc[0][1]);
    acc[0][2] = WMMA_F32(a0, b2, acc[0][2]);
    acc[0][3] = WMMA_F32(a0, b3, acc[0][3]);
    acc[1][0] = WMMA_F32(a1, b0, acc[1][0]);
    acc[1][1] = WMMA_F32(a1, b1, acc[1][1]);
    acc[1][2] = WMMA_F32(a1, b2, acc[1][2]);
    acc[1][3] = WMMA_F32(a1, b3, acc[1][3]);

    // C layout: VGPR r, lane l -> M = r + 8*lh, N = l16; store transposed.
#pragma unroll
    for (int tile = 0; tile < 2; ++tile) {
        int bb = bbase + tile * 16;
#pragma unroll
        for (int nt = 0; nt < 4; ++nt) {
            int n = nt * 16 + l16;
            float* outp = preT + ((size_t)(t * 64 + n)) * BATCH + bb + 8 * lh;
            v4f lo = {acc[tile][nt][0], acc[tile][nt][1], acc[tile][nt][2], acc[tile][nt][3]};
            v4f hi = {acc[tile][nt][4], acc[tile][nt][5], acc[tile][nt][6], acc[tile][nt][7]};
            *(v4f*)(outp + 0) = lo;
            *(v4f*)(outp + 4) = hi;
        }
    }
}

// ---------------------------------------------------------------------------
// Kernel C: 128-step recurrence, one thread per batch row, h/c in registers.
// ---------------------------------------------------------------------------
__device__ __forceinline__ float sigmoidf_(float x) {
    return 1.0f / (1.0f + __expf(-x));
}

__global__ __launch_bounds__(256) void qlstm_rec(
    const float* __restrict__ preT,    // [128][64][512]
    const float* __restrict__ Whpack,  // [64][8]
    float* __restrict__ out)           // outs(128*512*8) ++ h(4096) ++ c(4096)
{
    __shared__ float Wh[NCOL * NQ];
    for (int i = threadIdx.x; i < NCOL * NQ; i += blockDim.x) Wh[i] = Whpack[i];
    __syncthreads();

    int b = blockIdx.x * blockDim.x + threadIdx.x;
    if (b >= BATCH) return;

    float h[NQ], c[NQ];
#pragma unroll
    for (int q = 0; q < NQ; ++q) { h[q] = 0.f; c[q] = 0.f; }

    for (int t = 0; t < SEQ; ++t) {
        const float* pt = preT + (size_t)t * NCOL * BATCH + b;
        // pull next timestep's pre-activation lines toward the caches
        const float* pn = pt + (size_t)NCOL * BATCH;
#pragma unroll
        for (int col = 0; col < NCOL; col += 4)
            __builtin_prefetch(pn + (size_t)col * BATCH, 0, 0);

        float gates[4][NQ];
#pragma unroll
        for (int g = 0; g < 4; ++g) {
            float prod = 1.0f;
#pragma unroll
            for (int q = 0; q < NQ; ++q) {
                int col0 = g * 16 + q * 2;
                float a0 = pt[(size_t)col0 * BATCH];        // coalesced across b
                float a1 = pt[(size_t)(col0 + 1) * BATCH];
                const float* w0 = &Wh[col0 * NQ];
                const float* w1 = w0 + NQ;
#pragma unroll
                for (int j = 0; j < NQ; ++j) {
                    a0 = fmaf(h[j], w0[j], a0);
                    a1 = fmaf(h[j], w1[j], a1);
                }
                prod *= __cosf(a0) * __cosf(a1);
                gates[g][q] = prod;
            }
        }
#pragma unroll
        for (int q = 0; q < NQ; ++q) {
            float fg = sigmoidf_(gates[0][q]);
            float ig = sigmoidf_(gates[1][q]);
            float gg = tanhf(gates[2][q]);
            float og = sigmoidf_(gates[3][q]);
            c[q] = fg * c[q] + ig * gg;
            h[q] = og * tanhf(c[q]);
        }
        float* op = out + ((size_t)t * BATCH + b) * NQ;
        *(v4f*)(op + 0) = {h[0], h[1], h[2], h[3]};
        *(v4f*)(op + 4) = {h[4], h[5], h[6], h[7]};
    }

    float* hp = out + (size_t)SEQ * BATCH * NQ + b * NQ;
    float* cp = hp + BATCH * NQ;
    *(v4f*)(hp + 0) = {h[0], h[1], h[2], h[3]};
    *(v4f*)(hp + 4) = {h[4], h[5], h[6], h[7]};
    *(v4f*)(cp + 0) = {c[0], c[1], c[2], c[3]};
    *(v4f*)(cp + 4) = {c[4], c[5], c[6], c[7]};
}

// ---------------------------------------------------------------------------
extern "C" void kernel_launch(void* const* d_in, const int* in_sizes, int n_in,
                              void* d_out, int out_size, void* d_ws, size_t ws_size,
                              hipStream_t stream) {
    const float* X  = (const float*)d_in[0];
    const float* Wf = (const float*)d_in[1];
    const float* bf = (const float*)d_in[2];
    const float* Wi = (const float*)d_in[3];
    const float* bi = (const float*)d_in[4];
    const float* Wg = (const float*)d_in[5];
    const float* bg = (const float*)d_in[6];
    const float* Wo = (const float*)d_in[7];
    const float* bo = (const float*)d_in[8];
    float* out = (float*)d_out;

    char* base = (char*)d_ws;
    float* preT   = (float*)base;                                    // 16 MB
    float* Wpack  = (float*)(base + (size_t)SEQ * NCOL * BATCH * 4); // 128 KB
    float* Whpack = Wpack + BPANEL_ELEMS;                            // 2 KB
    float* bpack  = Whpack + NCOL * NQ;                              // 256 B

    {
        int total = NCOL * 256 + NCOL * NQ + NCOL;
        pack_weights<<<(total + 255) / 256, 256, 0, stream>>>(
            Wf, bf, Wi, bi, Wg, bg, Wo, bo, Wpack, Whpack, bpack);
    }
    {
        int blocks = (MTOT / 16) / (8 * 2);   // 2048 tile-pairs / 8 waves = 256 WGs
        qgemm<<<blocks, 256, BPANEL_ELEMS * sizeof(float), stream>>>(X, Wpack, bpack, preT);
    }
    {
        qlstm_rec<<<BATCH / 256, 256, 0, stream>>>(preT, Whpack, out);
    }
}